// PatientGNN_44263932952921
// MI455X (gfx1250) — compile-verified
//
#include <hip/hip_runtime.h>

typedef __attribute__((ext_vector_type(2))) float v2f;
typedef __attribute__((ext_vector_type(8))) float v8f;

#define TILE_K 64
#define NEG_SLOPE 0.2f
#define SOFT_EPS 1e-16f

// ---------------------------------------------------------------------------
// GEMM: C[Nrows,Ncols] = A[Nrows,K] * B[K,Ncols], row-major fp32, WMMA f32.
// Block = 128 threads (4 waves). Wave w computes rows [blk.x*64 + w*16, +16),
// cols [blk.y*COLS, +COLS) as COLS/16 WMMA 16x16 accumulators.
//
// B panel staged in LDS with K-pair interleaving:
//   Bs[(k>>1)*LSTRIDE + col*2 + (k&1)] = B[k][col]
// so each WMMA B fragment {B[k][col], B[k+1][col]} is a single aligned
// ds_load_b64. LSTRIDE = 2*COLS + 32 floats puts even pair-rows (lanes 0-15)
// on banks 0-31 and odd pair-rows (lanes 16-31) on banks 32-63: conflict-free.
//
// Requires: Ncols % COLS == 0, K % TILE_K == 0.
// ---------------------------------------------------------------------------
template <int COLS>
__global__ __launch_bounds__(128) void gatv2_gemm_wmma_f32(
    const float* __restrict__ A, const float* __restrict__ B,
    float* __restrict__ C, int Nrows, int K, int Ncols)
{
    constexpr int NACC    = COLS / 16;
    constexpr int LSTRIDE = 2 * COLS + 32;           // floats per K-pair row
    __shared__ float Bs[(TILE_K / 2) * LSTRIDE];

    const int tid  = threadIdx.x;
    const int lane = tid & 31;
    const int wave = tid >> 5;
    const int half = lane >> 4;       // 0: lanes 0-15, 1: lanes 16-31
    const int l15  = lane & 15;
    const int rowBase = blockIdx.x * 64 + wave * 16;
    const int colBase = blockIdx.y * COLS;

    v8f acc[NACC];
    #pragma unroll
    for (int j = 0; j < NACC; ++j) acc[j] = v8f{};

    // A row for this lane (clamp OOB rows so loads stay in-bounds; stores guarded)
    int arow = rowBase + l15;
    if (arow >= Nrows) arow = Nrows - 1;
    const float* __restrict__ Arow = A + (size_t)arow * K;

    // Staging geometry: each thread owns a 4-col group and strides over K-pairs.
    constexpr int CG  = COLS / 4;     // col groups per pair-row
    constexpr int RPS = 128 / CG;     // threads-worth of pair-rows per pass
    const int cg  = tid % CG;
    const int c   = cg * 4;
    const int rp0 = tid / CG;

    for (int k0 = 0; k0 < K; k0 += TILE_K) {
        // ---- stage B[k0..k0+TILE_K) x [colBase, +COLS) into LDS (interleaved)
        for (int rp = rp0; rp < TILE_K / 2; rp += RPS) {
            const float* __restrict__ b0 =
                B + (size_t)(k0 + 2 * rp) * Ncols + colBase + c;
            const float4 r0 = *(const float4*)(b0);
            const float4 r1 = *(const float4*)(b0 + Ncols);
            float* __restrict__ w = Bs + rp * LSTRIDE + c * 2;
            const float4 w0 = {r0.x, r1.x, r0.y, r1.y};
            const float4 w1 = {r0.z, r1.z, r0.w, r1.w};
            *(float4*)(w)     = w0;
            *(float4*)(w + 4) = w1;
        }
        // Prefetch next A tile while we wait at the barrier.
        if (k0 + TILE_K < K)
            __builtin_prefetch(Arow + k0 + TILE_K, 0, 0);
        __syncthreads();

        #pragma unroll 4
        for (int kk = 0; kk < TILE_K; kk += 4) {
            // A fragment 16x4: lanes 0-15 hold K=kk..kk+1, lanes 16-31 K=kk+2..kk+3
            const v2f a = *(const v2f*)(Arow + k0 + kk + (half << 1));
            // Pair-row: lanes 0-15 -> (kk>>1) [even], lanes 16-31 -> +1 [odd]
            const float* __restrict__ bp =
                Bs + ((kk >> 1) + half) * LSTRIDE + (l15 << 1);
            #pragma unroll
            for (int j = 0; j < NACC; ++j) {
                const v2f b = *(const v2f*)(bp + j * 32);
                acc[j] = __builtin_amdgcn_wmma_f32_16x16x4_f32(
                    false, a, false, b, (short)0, acc[j], false, false);
            }
        }
        __syncthreads();
    }

    // C layout: VGPR v -> row rowBase + v + half*8, col colBase + j*16 + l15
    const int crowBase = rowBase + half * 8;
    #pragma unroll
    for (int v = 0; v < 8; ++v) {
        const int r = crowBase + v;
        if (r < Nrows) {
            float* __restrict__ crow = C + (size_t)r * Ncols + colBase + l15;
            #pragma unroll
            for (int j = 0; j < NACC; ++j) crow[j * 16] = acc[j][v];
        }
    }
}

// Monotonic float<->uint mapping for atomicMax-based segment max.
__device__ __forceinline__ unsigned f32_to_key(float f) {
    unsigned u = __float_as_uint(f);
    return (u & 0x80000000u) ? ~u : (u | 0x80000000u);
}
__device__ __forceinline__ float key_to_f32(unsigned k) {
    unsigned u = (k & 0x80000000u) ? (k & 0x7FFFFFFFu) : ~k;
    return __uint_as_float(u);
}

// ---------------------------------------------------------------------------
// Layer-1 edge scoring: H=4, D=128, feature width 512. One wave per edge.
// logits[e,h] = sum_d att[h,d] * leakyrelu(xl[src,h,d] + xr[dst,h,d])
// Fused segment-max via atomicMax on encoded keys.
// ---------------------------------------------------------------------------
__global__ __launch_bounds__(256) void gatv2_edge_score1(
    const int* __restrict__ ei, const float* __restrict__ xl,
    const float* __restrict__ xr, const float* __restrict__ att,
    float* __restrict__ logits, unsigned* __restrict__ maxkey, int E, int N)
{
    const int e = blockIdx.x * 8 + (threadIdx.x >> 5);
    if (e >= E + N) return;
    const int lane = threadIdx.x & 31;
    const int s = (e < E) ? ei[e]     : (e - E);
    const int d = (e < E) ? ei[E + e] : (e - E);
    const float* __restrict__ pl = xl + (size_t)s * 512;
    const float* __restrict__ pr = xr + (size_t)d * 512;

    #pragma unroll
    for (int h = 0; h < 4; ++h) {
        const int off = h * 128 + lane * 4;
        const float4 a = *(const float4*)(pl + off);
        const float4 b = *(const float4*)(pr + off);
        const float4 w = *(const float4*)(att + off);
        float v0 = a.x + b.x; v0 = v0 > 0.0f ? v0 : NEG_SLOPE * v0;
        float v1 = a.y + b.y; v1 = v1 > 0.0f ? v1 : NEG_SLOPE * v1;
        float v2 = a.z + b.z; v2 = v2 > 0.0f ? v2 : NEG_SLOPE * v2;
        float v3 = a.w + b.w; v3 = v3 > 0.0f ? v3 : NEG_SLOPE * v3;
        float sum = v0 * w.x + v1 * w.y + v2 * w.z + v3 * w.w;
        #pragma unroll
        for (int m = 16; m > 0; m >>= 1) sum += __shfl_xor(sum, m, 32);
        if (lane == 0) {
            logits[(size_t)e * 4 + h] = sum;
            atomicMax(&maxkey[d * 4 + h], f32_to_key(sum));
        }
    }
}

// ---------------------------------------------------------------------------
// Softmax normalizer: ex = exp(logit - max[dst]); denom[dst] += ex. Generic H.
// ex written in-place over logits.
// ---------------------------------------------------------------------------
__global__ __launch_bounds__(256) void gatv2_softmax_norm(
    const int* __restrict__ ei, float* __restrict__ lg,
    const unsigned* __restrict__ maxkey, float* __restrict__ denom,
    int E, int N, int H)
{
    const long long idx = (long long)blockIdx.x * blockDim.x + threadIdx.x;
    if (idx >= (long long)(E + N) * H) return;
    const int e = (int)(idx / H);
    const int h = (int)(idx % H);
    const int d = (e < E) ? ei[E + e] : (e - E);
    const float m = key_to_f32(maxkey[d * H + h]);
    const float v = __expf(lg[idx] - m);
    lg[idx] = v;
    atomicAdd(&denom[d * H + h], v);
}

// ---------------------------------------------------------------------------
// Layer-1 aggregation: out[dst,h,d] += xl[src,h,d] * alpha(e,h). 128 thr/edge.
// ---------------------------------------------------------------------------
__global__ __launch_bounds__(256) void gatv2_aggregate1(
    const int* __restrict__ ei, const float* __restrict__ xl,
    const float* __restrict__ ex, const float* __restrict__ denom,
    float* __restrict__ out, int E, int N)
{
    const int e = blockIdx.x * 2 + (threadIdx.x >> 7);
    if (e >= E + N) return;
    const int t = threadIdx.x & 127;
    const int h = t >> 5;
    const int lane = t & 31;
    const int s = (e < E) ? ei[e]     : (e - E);
    const int d = (e < E) ? ei[E + e] : (e - E);
    const float alpha = ex[(size_t)e * 4 + h] / (denom[d * 4 + h] + SOFT_EPS);
    const int off = h * 128 + lane * 4;
    const float4 a = *(const float4*)(xl + (size_t)s * 512 + off);
    float* __restrict__ po = out + (size_t)d * 512 + off;
    atomicAdd(po + 0, a.x * alpha);
    atomicAdd(po + 1, a.y * alpha);
    atomicAdd(po + 2, a.z * alpha);
    atomicAdd(po + 3, a.w * alpha);
}

__global__ __launch_bounds__(256) void gatv2_bias_elu(
    float* __restrict__ p, const float* __restrict__ b, int total)
{
    const int i = blockIdx.x * blockDim.x + threadIdx.x;
    if (i >= total) return;
    const float v = p[i] + b[i & 511];
    p[i] = v > 0.0f ? v : (__expf(v) - 1.0f);
}

// ---------------------------------------------------------------------------
// Layer-2 edge scoring: H=1, D=64. One wave per edge, float2 per lane.
// ---------------------------------------------------------------------------
__global__ __launch_bounds__(256) void gatv2_edge_score2(
    const int* __restrict__ ei, const float* __restrict__ xl,
    const float* __restrict__ xr, const float* __restrict__ att,
    float* __restrict__ logits, unsigned* __restrict__ maxkey, int E, int N)
{
    const int e = blockIdx.x * 8 + (threadIdx.x >> 5);
    if (e >= E + N) return;
    const int lane = threadIdx.x & 31;
    const int s = (e < E) ? ei[e]     : (e - E);
    const int d = (e < E) ? ei[E + e] : (e - E);
    const int off = lane * 2;
    const float2 a = *(const float2*)(xl + (size_t)s * 64 + off);
    const float2 b = *(const float2*)(xr + (size_t)d * 64 + off);
    const float2 w = *(const float2*)(att + off);
    float v0 = a.x + b.x; v0 = v0 > 0.0f ? v0 : NEG_SLOPE * v0;
    float v1 = a.y + b.y; v1 = v1 > 0.0f ? v1 : NEG_SLOPE * v1;
    float sum = v0 * w.x + v1 * w.y;
    #pragma unroll
    for (int m = 16; m > 0; m >>= 1) sum += __shfl_xor(sum, m, 32);
    if (lane == 0) {
        logits[e] = sum;
        atomicMax(&maxkey[d], f32_to_key(sum));
    }
}

// ---------------------------------------------------------------------------
// Layer-2 aggregation into d_out (64 features). One wave per edge.
// ---------------------------------------------------------------------------
__global__ __launch_bounds__(256) void gatv2_aggregate2(
    const int* __restrict__ ei, const float* __restrict__ xl,
    const float* __restrict__ ex, const float* __restrict__ denom,
    float* __restrict__ out, int E, int N)
{
    const int e = blockIdx.x * 8 + (threadIdx.x >> 5);
    if (e >= E + N) return;
    const int lane = threadIdx.x & 31;
    const int s = (e < E) ? ei[e]     : (e - E);
    const int d = (e < E) ? ei[E + e] : (e - E);
    const float alpha = ex[e] / (denom[d] + SOFT_EPS);
    const int off = lane * 2;
    const float2 a = *(const float2*)(xl + (size_t)s * 64 + off);
    float* __restrict__ po = out + (size_t)d * 64 + off;
    atomicAdd(po + 0, a.x * alpha);
    atomicAdd(po + 1, a.y * alpha);
}

__global__ __launch_bounds__(256) void gatv2_bias_add(
    float* __restrict__ p, const float* __restrict__ b, int mod, int total)
{
    const int i = blockIdx.x * blockDim.x + threadIdx.x;
    if (i >= total) return;
    p[i] += b[i % mod];
}

// ---------------------------------------------------------------------------
extern "C" void kernel_launch(void* const* d_in, const int* in_sizes, int n_in,
                              void* d_out, int out_size, void* d_ws, size_t ws_size,
                              hipStream_t stream) {
    const float* x    = (const float*)d_in[0];
    const int*   ei   = (const int*)  d_in[1];   // [2,E] int32
    const float* Wl1  = (const float*)d_in[2];   // [256,512]
    const float* Wr1  = (const float*)d_in[3];
    const float* att1 = (const float*)d_in[4];   // [4,128]
    const float* b1   = (const float*)d_in[5];   // [512]
    const float* Wl2  = (const float*)d_in[6];   // [512,64]
    const float* Wr2  = (const float*)d_in[7];
    const float* att2 = (const float*)d_in[8];   // [64]
    const float* b2   = (const float*)d_in[9];   // [64]
    float* out = (float*)d_out;

    const int N  = in_sizes[0] / 256;   // 50000
    const int E  = in_sizes[1] / 2;     // 400000
    const int ET = E + N;               // with self-loops
    const size_t NHD = (size_t)N * 512;

    // Workspace layout (floats); layer-2 buffers overlay dead layer-1 buffers.
    float* ws = (float*)d_ws;
    float*    xl1  = ws;                         // N*512
    float*    xr1  = ws + NHD;                   // N*512
    float*    hbuf = ws + 2 * NHD;               // N*512
    float*    lg1  = ws + 3 * NHD;               // ET*4 (logits, then ex in-place)
    unsigned* mk1  = (unsigned*)(ws + 3 * NHD + (size_t)ET * 4); // N*4
    float*    dn1  = (float*)(mk1 + (size_t)N * 4);              // N*4
    // layer-2 overlays
    float*    xl2 = xl1;            // N*64
    float*    xr2 = xr1;            // N*64
    float*    lg2 = lg1;            // ET
    unsigned* mk2 = mk1;            // N
    float*    dn2 = dn1;            // N

    // ---- Layer 1 ----
    dim3 ggrid1((N + 63) / 64, 512 / 128);
    gatv2_gemm_wmma_f32<128><<<ggrid1, 128, 0, stream>>>(x, Wl1, xl1, N, 256, 512);
    gatv2_gemm_wmma_f32<128><<<ggrid1, 128, 0, stream>>>(x, Wr1, xr1, N, 256, 512);

    hipMemsetAsync(mk1,  0, (size_t)N * 4 * sizeof(unsigned), stream);
    hipMemsetAsync(dn1,  0, (size_t)N * 4 * sizeof(float), stream);
    hipMemsetAsync(hbuf, 0, NHD * sizeof(float), stream);

    gatv2_edge_score1<<<(ET + 7) / 8, 256, 0, stream>>>(ei, xl1, xr1, att1, lg1, mk1, E, N);
    {
        long long tot = (long long)ET * 4;
        gatv2_softmax_norm<<<(unsigned)((tot + 255) / 256), 256, 0, stream>>>(
            ei, lg1, mk1, dn1, E, N, 4);
    }
    gatv2_aggregate1<<<(ET + 1) / 2, 256, 0, stream>>>(ei, xl1, lg1, dn1, hbuf, E, N);
    gatv2_bias_elu<<<(int)((NHD + 255) / 256), 256, 0, stream>>>(hbuf, b1, (int)NHD);

    // ---- Layer 2 ----
    dim3 ggrid2((N + 63) / 64, 1);
    gatv2_gemm_wmma_f32<64><<<ggrid2, 128, 0, stream>>>(hbuf, Wl2, xl2, N, 512, 64);
    gatv2_gemm_wmma_f32<64><<<ggrid2, 128, 0, stream>>>(hbuf, Wr2, xr2, N, 512, 64);

    hipMemsetAsync(mk2, 0, (size_t)N * sizeof(unsigned), stream);
    hipMemsetAsync(dn2, 0, (size_t)N * sizeof(float), stream);
    hipMemsetAsync(out, 0, (size_t)N * 64 * sizeof(float), stream);

    gatv2_edge_score2<<<(ET + 7) / 8, 256, 0, stream>>>(ei, xl2, xr2, att2, lg2, mk2, E, N);
    gatv2_softmax_norm<<<(ET + 255) / 256, 256, 0, stream>>>(ei, lg2, mk2, dn2, E, N, 1);
    gatv2_aggregate2<<<(ET + 7) / 8, 256, 0, stream>>>(ei, xl2, lg2, dn2, out, E, N);
    gatv2_bias_add<<<(N * 64 + 255) / 256, 256, 0, stream>>>(out, b2, 64, N * 64);

    (void)n_in; (void)out_size; (void)ws_size;
}